// TopKCrossEntropyLoss_63161789055480
// MI455X (gfx1250) — compile-verified
//
#include <hip/hip_runtime.h>
#include <math.h>

#define B_SZ 8
#define C_CLS 19
#define HW (512 * 512)
#define K_SEL 52429            // round(0.2 * 262144)
#define NBINS 65536
#define IGNORE_LABEL 255

// ---- workspace layout (bytes) ----
static constexpr size_t OFF_LOSS   = 0;                                   // 8 * HW * 4 = 8 MB
static constexpr size_t OFF_HISTHI = (size_t)B_SZ * HW * 4;               // 8 * 65536 * 4 = 2 MB
static constexpr size_t OFF_HISTLO = OFF_HISTHI + (size_t)B_SZ * NBINS * 4;
static constexpr size_t OFF_SUMLO  = OFF_HISTLO + (size_t)B_SZ * NBINS * 4;
static constexpr size_t OFF_MISC   = OFF_SUMLO  + (size_t)B_SZ * NBINS * 4;

struct Misc {
    unsigned binHi[B_SZ];     // high-16-bit bin containing the k-th largest
    unsigned kRem[B_SZ];      // elements still needed from that bin
    double   sumAbove[B_SZ];  // sum of values with hi-bin > binHi
    double   sampleSum[B_SZ]; // final per-sample top-k sum
};

// ------------------------------------------------------------------
// Zero scratch (histograms + control block)
// ------------------------------------------------------------------
__global__ __launch_bounds__(256) void zero_kernel(unsigned* __restrict__ p, size_t n) {
    size_t i = (size_t)blockIdx.x * blockDim.x + threadIdx.x;
    size_t stride = (size_t)gridDim.x * blockDim.x;
    for (; i < n; i += stride) p[i] = 0u;
}

// ------------------------------------------------------------------
// Pass 1: weighted NLL per pixel (4 pixels/thread) + hi-16 histogram
// ------------------------------------------------------------------
__global__ __launch_bounds__(256) void loss_hist_kernel(const int* __restrict__ labels,
                                                        const float* __restrict__ logits,
                                                        const float* __restrict__ weights,
                                                        float* __restrict__ loss,
                                                        unsigned* __restrict__ histHi) {
    const int tid = blockIdx.x * blockDim.x + threadIdx.x;  // one thread = 4 pixels
    const int p4  = tid * 4;
    const int b   = p4 / HW;                 // HW % 4 == 0: never straddles a sample
    const int p   = p4 - b * HW;

    const float* base = logits + (size_t)b * C_CLS * HW + p;
    float x[4][C_CLS];
#pragma unroll
    for (int c = 0; c < C_CLS; ++c) {
        float4 v = *reinterpret_cast<const float4*>(base + (size_t)c * HW);
        x[0][c] = v.x; x[1][c] = v.y; x[2][c] = v.z; x[3][c] = v.w;
    }
    int4   lab4 = *reinterpret_cast<const int4*>(labels + (size_t)b * HW + p);
    float4 w4   = *reinterpret_cast<const float4*>(weights + (size_t)b * HW + p);
    const int   labs[4] = {lab4.x, lab4.y, lab4.z, lab4.w};
    const float ws[4]   = {w4.x, w4.y, w4.z, w4.w};

    float out[4];
#pragma unroll
    for (int j = 0; j < 4; ++j) {
        float m = -INFINITY;
#pragma unroll
        for (int c = 0; c < C_CLS; ++c) m = fmaxf(m, x[j][c]);
        float s = 0.0f, xl = 0.0f;
#pragma unroll
        for (int c = 0; c < C_CLS; ++c) {
            float v = x[j][c];
            s += expf(v - m);
            if (c == labs[j]) xl = v;   // safe: if label==IGNORE no c matches
        }
        float nll = logf(s) + m - xl;
        float l   = (labs[j] != IGNORE_LABEL) ? nll * ws[j] : 0.0f;
        if (!(l > 0.0f)) l = 0.0f;      // clamp tiny negatives / NaN -> 0 (monotone u32 bits)
        out[j] = l;
        unsigned bits = __float_as_uint(l);
        atomicAdd(&histHi[(size_t)b * NBINS + (bits >> 16)], 1u);
    }
    *reinterpret_cast<float4*>(loss + (size_t)b * HW + p) =
        make_float4(out[0], out[1], out[2], out[3]);
}

// ------------------------------------------------------------------
// Pass 2: per-sample suffix scan of hi histogram -> binHi, kRem
// One 256-thread block per sample; thread t owns bins [t*256, t*256+256)
// ------------------------------------------------------------------
__global__ __launch_bounds__(256) void scan_hi_kernel(const unsigned* __restrict__ histHi,
                                                      Misc* __restrict__ misc) {
    const int b = blockIdx.x;
    const int t = threadIdx.x;
    const unsigned* h = histHi + (size_t)b * NBINS;

    unsigned chunk = 0;
    const uint4* h4 = reinterpret_cast<const uint4*>(h + t * 256);
#pragma unroll 8
    for (int i = 0; i < 64; ++i) {
        uint4 u = h4[i];
        chunk += u.x + u.y + u.z + u.w;
    }
    __shared__ unsigned s[256];
    s[t] = chunk;
    __syncthreads();
    for (int off = 1; off < 256; off <<= 1) {          // inclusive suffix scan
        unsigned v = (t + off < 256) ? s[t + off] : 0u;
        __syncthreads();
        s[t] += v;
        __syncthreads();
    }
    const unsigned cumAbove = (t < 255) ? s[t + 1] : 0u;  // count strictly above my chunk
    if (cumAbove < K_SEL && cumAbove + chunk >= K_SEL) {  // exactly one thread
        unsigned running = cumAbove;
        for (int i = 255; i >= 0; --i) {
            unsigned c = h[t * 256 + i];
            if (running + c >= K_SEL) {
                misc->binHi[b] = (unsigned)(t * 256 + i);
                misc->kRem[b]  = K_SEL - running;
                break;
            }
            running += c;
        }
    }
}

// ------------------------------------------------------------------
// Pass 3: sum(values in bins > binHi) [block-reduced f64 atomics] and
// lo-16 count/sum histograms for values inside bin binHi
// ------------------------------------------------------------------
__global__ __launch_bounds__(256) void lowhist_kernel(const float* __restrict__ loss,
                                                      unsigned* __restrict__ histLo,
                                                      float* __restrict__ sumLo,
                                                      Misc* __restrict__ misc) {
    const int blocksPerSample = HW / 1024;           // 256
    const int b    = blockIdx.x / blocksPerSample;
    const int pblk = (blockIdx.x % blocksPerSample) * 1024;
    const unsigned binHi = misc->binHi[b];           // uniform -> scalar load

    const int p = pblk + threadIdx.x * 4;
    float4 v = *reinterpret_cast<const float4*>(loss + (size_t)b * HW + p);
    const float vv[4] = {v.x, v.y, v.z, v.w};

    double local = 0.0;
#pragma unroll
    for (int j = 0; j < 4; ++j) {
        unsigned bits = __float_as_uint(vv[j]);
        unsigned hi = bits >> 16;
        if (hi > binHi) {
            local += (double)vv[j];
        } else if (hi == binHi) {
            unsigned lo = bits & 0xFFFFu;
            atomicAdd(&histLo[(size_t)b * NBINS + lo], 1u);
            atomicAdd(&sumLo[(size_t)b * NBINS + lo], vv[j]);   // bit-identical values: order-safe
        }
    }
    __shared__ double red[256];
    red[threadIdx.x] = local;
    __syncthreads();
    for (int s = 128; s > 0; s >>= 1) {
        if (threadIdx.x < s) red[threadIdx.x] += red[threadIdx.x + s];
        __syncthreads();
    }
    if (threadIdx.x == 0 && red[0] != 0.0)
        atomicAdd(&misc->sumAbove[b], red[0]);
}

// ------------------------------------------------------------------
// Pass 4: suffix scan of lo histogram -> exact k-th value, per-sample top-k sum
// ------------------------------------------------------------------
__global__ __launch_bounds__(256) void scan_lo_kernel(const unsigned* __restrict__ histLo,
                                                      const float* __restrict__ sumLo,
                                                      Misc* __restrict__ misc) {
    const int b = blockIdx.x;
    const int t = threadIdx.x;
    const unsigned* h  = histLo + (size_t)b * NBINS;
    const float*    sm = sumLo  + (size_t)b * NBINS;
    const unsigned kRem  = misc->kRem[b];
    const unsigned binHi = misc->binHi[b];

    unsigned ccnt = 0;
    double   csum = 0.0;
    for (int i = 0; i < 256; ++i) {
        unsigned c = h[t * 256 + i];
        ccnt += c;
        if (c) csum += (double)sm[t * 256 + i];
    }
    __shared__ unsigned scnt[256];
    __shared__ double   ssum[256];
    scnt[t] = ccnt; ssum[t] = csum;
    __syncthreads();
    for (int off = 1; off < 256; off <<= 1) {
        unsigned vc = (t + off < 256) ? scnt[t + off] : 0u;
        double   vs = (t + off < 256) ? ssum[t + off] : 0.0;
        __syncthreads();
        scnt[t] += vc; ssum[t] += vs;
        __syncthreads();
    }
    const unsigned cumAboveC = (t < 255) ? scnt[t + 1] : 0u;
    const double   cumAboveS = (t < 255) ? ssum[t + 1] : 0.0;
    if (cumAboveC < kRem && cumAboveC + ccnt >= kRem) {   // exactly one thread
        unsigned running = cumAboveC;
        double   runSum  = cumAboveS;
        for (int i = 255; i >= 0; --i) {
            unsigned c = h[t * 256 + i];
            if (running + c >= kRem) {
                unsigned bits = (binHi << 16) | (unsigned)(t * 256 + i);
                float tval = __uint_as_float(bits);        // exact k-th largest value
                misc->sampleSum[b] =
                    misc->sumAbove[b] + runSum + (double)(kRem - running) * (double)tval;
                break;
            }
            running += c;
            runSum  += (double)sm[t * 256 + i];
        }
    }
}

// ------------------------------------------------------------------
// Pass 5: mean over B*k, scale by LOSS_WEIGHT (=1)
// ------------------------------------------------------------------
__global__ void finalize_kernel(const Misc* __restrict__ misc, float* __restrict__ out) {
    if (threadIdx.x == 0 && blockIdx.x == 0) {
        double s = 0.0;
        for (int b = 0; b < B_SZ; ++b) s += misc->sampleSum[b];
        out[0] = (float)(s / ((double)B_SZ * (double)K_SEL));
    }
}

extern "C" void kernel_launch(void* const* d_in, const int* in_sizes, int n_in,
                              void* d_out, int out_size, void* d_ws, size_t ws_size,
                              hipStream_t stream) {
    const int*   y_true  = (const int*)d_in[0];
    const float* y_pred  = (const float*)d_in[1];
    const float* weights = (const float*)d_in[2];
    float* out = (float*)d_out;

    char* ws = (char*)d_ws;
    float*    loss   = (float*)(ws + OFF_LOSS);
    unsigned* histHi = (unsigned*)(ws + OFF_HISTHI);
    unsigned* histLo = (unsigned*)(ws + OFF_HISTLO);
    float*    sumLo  = (float*)(ws + OFF_SUMLO);
    Misc*     misc   = (Misc*)(ws + OFF_MISC);

    const size_t zwords = (OFF_MISC + sizeof(Misc) - OFF_HISTHI) / 4;
    zero_kernel<<<2048, 256, 0, stream>>>((unsigned*)(ws + OFF_HISTHI), zwords);

    loss_hist_kernel<<<(B_SZ * HW / 4) / 256, 256, 0, stream>>>(y_true, y_pred, weights,
                                                                loss, histHi);
    scan_hi_kernel<<<B_SZ, 256, 0, stream>>>(histHi, misc);
    lowhist_kernel<<<B_SZ * (HW / 1024), 256, 0, stream>>>(loss, histLo, sumLo, misc);
    scan_lo_kernel<<<B_SZ, 256, 0, stream>>>(histLo, sumLo, misc);
    finalize_kernel<<<1, 32, 0, stream>>>(misc, out);
}